// ReLU_80942953660923
// MI455X (gfx1250) — compile-verified
//
#include <hip/hip_runtime.h>

// Elementwise degree-7 polynomial (ReLU LSQ fit), Horner's rule.
// Memory-bound streaming: 1.07 GB traffic, ~46us floor at 23.3 TB/s.
// gfx1250 path: double-buffered GLOBAL_LOAD_ASYNC_TO_LDS_B128 pipeline,
// exact S_WAIT_ASYNCCNT sync, NT cache policy on both streams,
// v_pk_fma_f32-friendly Horner chain.

#ifndef __has_builtin
#define __has_builtin(x) 0
#endif

#if __has_builtin(__builtin_amdgcn_global_load_async_to_lds_b128) && \
    __has_builtin(__builtin_amdgcn_s_wait_asynccnt)
#define USE_ASYNC_LDS 1
#else
#define USE_ASYNC_LDS 0
#endif

#define NCOEF 8
#define TPB 256
#define VEC 2          // float4s per lane per pipeline stage
#define CPOL_NT 1      // TH[2:0]=1 -> non-temporal (read-once stream)

// Native vector types (HIP's float4 is a class; builtins need real vectors).
typedef float f32x4 __attribute__((ext_vector_type(4)));
typedef int v4i __attribute__((vector_size(16)));
typedef __attribute__((address_space(1))) v4i gv4i;  // global
typedef __attribute__((address_space(3))) v4i lv4i;  // LDS

__device__ __forceinline__ float horner1(float x, const float c[NCOEF]) {
  float a = c[NCOEF - 1];
#pragma unroll
  for (int k = NCOEF - 2; k >= 0; --k) a = __builtin_fmaf(a, x, c[k]);
  return a;
}

__device__ __forceinline__ f32x4 horner4(f32x4 v, const float c[NCOEF]) {
  f32x4 r;
  r.x = horner1(v.x, c);
  r.y = horner1(v.y, c);
  r.z = horner1(v.z, c);
  r.w = horner1(v.w, c);
  return r;
}

__global__ __launch_bounds__(TPB) void poly_relu_kernel(
    const float* __restrict__ x, const float* __restrict__ coef,
    float* __restrict__ out, long long n) {
  // Broadcast coefficients (uniform address, hits WGP$/L2; negligible).
  float c[NCOEF];
#pragma unroll
  for (int k = 0; k < NCOEF; ++k) c[k] = coef[k];

  const long long n4 = n >> 2;                 // float4 chunks
  const long long tilesz = (long long)TPB * VEC;  // float4s per tile
  const long long ntiles = n4 / tilesz;        // full tiles (EXEC-uniform)
  const long long tstride = gridDim.x;
  const f32x4* __restrict__ x4 = (const f32x4*)x;
  f32x4* __restrict__ o4 = (f32x4*)out;
  const int t = threadIdx.x;

#if USE_ASYNC_LDS
  // Double-buffered async DMA into LDS: each lane owns its slots, so the
  // per-wave S_WAIT_ASYNCCNT is the only synchronization (no barriers).
  __shared__ f32x4 lbuf[2][VEC][TPB];
  long long tile = blockIdx.x;
  int cur = 0;
  if (tile < ntiles) {
    const f32x4* p = x4 + tile * tilesz + t;
#pragma unroll
    for (int j = 0; j < VEC; ++j)
      __builtin_amdgcn_global_load_async_to_lds_b128(
          (gv4i*)(p + j * TPB), (lv4i*)&lbuf[0][j][t], 0, CPOL_NT);
  }
  for (; tile < ntiles; tile += tstride) {
    const long long tnext = tile + tstride;
    if (tnext < ntiles) {
      // Prefetch next tile into the other buffer, then wait only for the
      // older stage (ASYNCcnt <= VEC): compute overlaps the in-flight DMA.
      const f32x4* p = x4 + tnext * tilesz + t;
#pragma unroll
      for (int j = 0; j < VEC; ++j)
        __builtin_amdgcn_global_load_async_to_lds_b128(
            (gv4i*)(p + j * TPB), (lv4i*)&lbuf[cur ^ 1][j][t], 0, CPOL_NT);
      __builtin_amdgcn_s_wait_asynccnt(VEC);
    } else {
      __builtin_amdgcn_s_wait_asynccnt(0);
    }
    f32x4* op = o4 + tile * tilesz + t;
#pragma unroll
    for (int j = 0; j < VEC; ++j) {
      f32x4 v = lbuf[cur][j][t];                              // ds_load_b128
      __builtin_nontemporal_store(horner4(v, c), op + j * TPB);  // NT b128
    }
    cur ^= 1;
  }
  // Remainder float4s (< tilesz): tiny, block 0 handles with plain NT loads.
  if (blockIdx.x == 0) {
    for (long long i = ntiles * tilesz + t; i < n4; i += TPB) {
      f32x4 v = __builtin_nontemporal_load(x4 + i);
      __builtin_nontemporal_store(horner4(v, c), o4 + i);
    }
  }
#else
  // Fallback: plain NT 128-bit streaming, grid-stride.
  for (long long i = (long long)blockIdx.x * TPB + t; i < n4;
       i += tstride * TPB) {
    f32x4 v = __builtin_nontemporal_load(x4 + i);
    __builtin_nontemporal_store(horner4(v, c), o4 + i);
  }
#endif

  // Scalar tail (n % 4 elements).
  const long long tail = n & 3LL;
  if (tail && blockIdx.x == 0 && t < (int)tail) {
    const long long j = n - tail + t;
    out[j] = horner1(x[j], c);
  }
}

extern "C" void kernel_launch(void* const* d_in, const int* in_sizes, int n_in,
                              void* d_out, int out_size, void* d_ws, size_t ws_size,
                              hipStream_t stream) {
  const float* x = (const float*)d_in[0];     // [32,4096,1024] fp32
  const float* coef = (const float*)d_in[1];  // [8] fp32
  float* out = (float*)d_out;
  const long long n = (long long)out_size;

  const long long n4 = (n + 3) >> 2;
  const long long ntiles = n4 / ((long long)TPB * VEC);
  long long want = ntiles > 0 ? ntiles : 1;
  int blocks = (int)(want < 4096 ? want : 4096);
  poly_relu_kernel<<<blocks, TPB, 0, stream>>>(x, coef, out, n);
}